// DotProductAttention_52398601012060
// MI455X (gfx1250) — compile-verified
//
#include <hip/hip_runtime.h>

typedef __attribute__((ext_vector_type(2))) float v2f;
typedef __attribute__((ext_vector_type(4))) float v4f;
typedef __attribute__((ext_vector_type(8))) float v8f;
typedef int v4i __attribute__((vector_size(4 * sizeof(int))));

// D = A(16x4) * B(4x16) + C(16x16), fp32. 8-arg form:
// (neg_a, A, neg_b, B, c_mod, C, reuse_a, reuse_b)
__device__ __forceinline__ v8f wmma_f32(v2f a, v2f b, v8f c) {
    return __builtin_amdgcn_wmma_f32_16x16x4_f32(false, a, false, b, (short)0, c, false, false);
}

constexpr int Bc = 2, Hc = 16, Lc = 2048, Dc = 64;
constexpr int MWAVE  = 16;   // query rows per wave
constexpr int WAVES  = 8;    // waves per block
constexpr int MBLOCK = MWAVE * WAVES;   // 128 query rows per block
constexpr int KBLOCK = 64;   // keys staged in LDS per outer iter
constexpr int KSUB   = 16;   // keys per WMMA sub-block
constexpr int KP = 68;       // LDS row pitch (floats) for K/V tiles (mult. of 4, conflict-free b64 reads)
constexpr int SP = 20;       // LDS row pitch (floats) for per-wave S scratch

#if defined(__HIP_DEVICE_COMPILE__) && __has_builtin(__builtin_amdgcn_global_load_async_to_lds_b128)
#define HAS_ASYNC_LDS 1
#else
#define HAS_ASYNC_LDS 0
#endif

// Copy 16 bytes global -> LDS. Async path uses GLOBAL_LOAD_ASYNC_TO_LDS_B128
// (tracked with ASYNCcnt); fallback is a plain load+store (tracked with
// LOADcnt/DScnt, ordered by the per-iteration barrier).
__device__ __forceinline__ void copy16_g2l(const float* __restrict__ g, float* l) {
#if HAS_ASYNC_LDS
    __builtin_amdgcn_global_load_async_to_lds_b128((v4i*)g, (v4i*)l, 0, 0);
#else
    *(v4f*)l = *(const v4f*)g;
#endif
}

__device__ __forceinline__ void wait_async_done() {
#if HAS_ASYNC_LDS
#if __has_builtin(__builtin_amdgcn_s_wait_asynccnt)
    __builtin_amdgcn_s_wait_asynccnt(0);
#else
    asm volatile("s_wait_asynccnt 0x0" ::: "memory");
#endif
#endif
}

__global__ __launch_bounds__(256)
void attn_qkv_f32_wmma(const float* __restrict__ Q,
                       const float* __restrict__ K,
                       const float* __restrict__ V,
                       float* __restrict__ O) {
    __shared__ float lK[2][KBLOCK * KP];
    __shared__ float lV[2][KBLOCK * KP];
    __shared__ float lS[WAVES * 16 * SP];

    const int tid  = threadIdx.x;
    const int wave = tid >> 5;
    const int lane = tid & 31;
    const int lm   = lane & 15;   // row (A/C) or col (B/C) index within 16
    const int half = lane >> 4;   // lane-half: selects K pair (A/B) or +8 rows (C/D)

    constexpr int nQB = Lc / MBLOCK;              // 16 query blocks per (b,h)
    const int bh = blockIdx.x / nQB;
    const int qb = blockIdx.x % nQB;

    const float* Qb = Q + (size_t)bh * Lc * Dc;
    const float* Kb = K + (size_t)bh * Lc * Dc;
    const float* Vb = V + (size_t)bh * Lc * Dc;
    float*       Ob = O + (size_t)bh * Lc * Dc;

    // Per-thread staging coordinates: 256 threads cover a 64x64 f32 tile,
    // 4 threads per row, 16 floats (four b128s) per thread; coalesced b128s.
    const int sr = tid >> 2;          // 0..63 (tile row)
    const int sc = (tid & 3) * 16;    // 0,16,32,48 (tile col)

    // ---- Load Q tile into A-layout registers: step s covers d = 4s..4s+3.
    // A 16x4 f32 layout: lanes 0-15 hold K=0,1 (VGPR0,1); lanes 16-31 hold K=2,3.
    const int qrow = qb * MBLOCK + wave * MWAVE + lm;
    v2f qa[Dc / 4];
#pragma unroll
    for (int s = 0; s < Dc / 4; ++s)
        qa[s] = *(const v2f*)(Qb + (size_t)qrow * Dc + 4 * s + 2 * half);

    v8f acc[4] = {};                       // O tile: 4 d-chunks of 16 cols, C layout
    float* sw = lS + wave * 16 * SP;       // per-wave S re-layout scratch

    // ---- Prologue: start async stage of the first K/V tile into buffer 0.
#pragma unroll
    for (int i = 0; i < 4; ++i) {
        copy16_g2l(Kb + (size_t)sr * Dc + sc + 4 * i, &lK[0][sr * KP + sc + 4 * i]);
        copy16_g2l(Vb + (size_t)sr * Dc + sc + 4 * i, &lV[0][sr * KP + sc + 4 * i]);
    }

    int p = 0;
    for (int kb = 0; kb < Lc; kb += KBLOCK, p ^= 1) {
        // Wait for this iteration's tile (issued last iteration) to land,
        // then one barrier: publishes buf[p] to all waves AND guarantees all
        // waves finished reading buf[p^1] before we overwrite it below.
        wait_async_done();
        __syncthreads();

        // ---- Prefetch next tile into the other buffer (overlaps with compute).
        if (kb + KBLOCK < Lc) {
            const int nb = kb + KBLOCK;
#pragma unroll
            for (int i = 0; i < 4; ++i) {
                copy16_g2l(Kb + (size_t)(nb + sr) * Dc + sc + 4 * i,
                           &lK[p ^ 1][sr * KP + sc + 4 * i]);
                copy16_g2l(Vb + (size_t)(nb + sr) * Dc + sc + 4 * i,
                           &lV[p ^ 1][sr * KP + sc + 4 * i]);
            }
        }

        const float* tK = lK[p];
        const float* tV = lV[p];

#pragma unroll
        for (int t = 0; t < KBLOCK / KSUB; ++t) {
            const int kt = t * KSUB;

            // ---- S = Q_tile @ K_tile^T : contract D via 16 k-steps of 4.
            // B 4x16 layout: lane = key col n; lanes 0-15 hold K=0,1; 16-31 hold K=2,3.
            v8f s = {};
#pragma unroll
            for (int ss = 0; ss < Dc / 4; ++ss) {
                v2f bk = *(const v2f*)(tK + (kt + lm) * KP + 4 * ss + 2 * half);
                s = wmma_f32(qa[ss], bk, s);
            }

            // ---- Re-layout S: C/D layout -> A layout via per-wave LDS scratch.
            // DS ops are processed in-order within a wave, so the wave's store
            // instructions complete before its subsequent load instructions.
#pragma unroll
            for (int r = 0; r < 8; ++r)
                sw[(r + 8 * half) * SP + lm] = s[r];
            asm volatile("" ::: "memory");   // pin compiler ordering across scratch
            v2f sa[4];
#pragma unroll
            for (int j = 0; j < 4; ++j)
                sa[j] = *(const v2f*)(sw + lm * SP + 4 * j + 2 * half);
            asm volatile("" ::: "memory");

            // ---- O += S_tile @ V_tile : contract 16 keys via 4 chunks of 4.
#pragma unroll
            for (int j = 0; j < 4; ++j) {
                const int krow = kt + 4 * j + 2 * half;  // B operand key rows (pair)
#pragma unroll
                for (int tc = 0; tc < 4; ++tc) {
                    v2f bv;
                    bv.x = tV[(krow + 0) * KP + tc * 16 + lm];
                    bv.y = tV[(krow + 1) * KP + tc * 16 + lm];
                    acc[tc] = wmma_f32(sa[j], bv, acc[tc]);
                }
            }
        }
        // No trailing barrier: the top-of-loop barrier provides the ordering.
    }

    // ---- Write O tile (C layout: VGPR r -> row r + 8*half, lane -> col).
    const int orow = qb * MBLOCK + wave * MWAVE + 8 * half;
#pragma unroll
    for (int tc = 0; tc < 4; ++tc)
#pragma unroll
        for (int r = 0; r < 8; ++r)
            Ob[(size_t)(orow + r) * Dc + tc * 16 + lm] = acc[tc][r];
}

extern "C" void kernel_launch(void* const* d_in, const int* in_sizes, int n_in,
                              void* d_out, int out_size, void* d_ws, size_t ws_size,
                              hipStream_t stream) {
    (void)in_sizes; (void)n_in; (void)out_size; (void)d_ws; (void)ws_size;
    const float* q = (const float*)d_in[0];
    const float* k = (const float*)d_in[1];
    const float* v = (const float*)d_in[2];
    float*       o = (float*)d_out;

    dim3 grid(Bc * Hc * (Lc / MBLOCK));   // 512 blocks
    dim3 block(WAVES * 32);               // 256 threads = 8 waves
    attn_qkv_f32_wmma<<<grid, block, 0, stream>>>(q, k, v, o);
}